// BahdanauAttnDecoderRNN_74242804679160
// MI455X (gfx1250) — compile-verified
//
#include <hip/hip_runtime.h>

#define Hdim   1024
#define EMBdim 512
#define SEQ    4096
#define NVOCAB 50257

typedef __attribute__((ext_vector_type(2))) float v2f;
typedef __attribute__((ext_vector_type(8))) float v8f;

// ---------------------------------------------------------------------------
// u[k] = sum_h v[h] * attn_W[h, Hdim + k]   (the "We^T v" collapse vector)
// ---------------------------------------------------------------------------
__global__ void k_u(const float* __restrict__ attn_W, const float* __restrict__ v,
                    float* __restrict__ u) {
    int k = blockIdx.x * blockDim.x + threadIdx.x;      // 0..Hdim-1
    float acc = 0.f;
    for (int h = 0; h < Hdim; ++h)
        acc += v[h] * attn_W[(size_t)h * (2 * Hdim) + Hdim + k];  // coalesced over k
    u[k] = acc;
}

// ---------------------------------------------------------------------------
// Generic fp32 GEMV on the WMMA pipe:  y[r] = A[r,:].x + bias[r]
//   - one wave per 16-row tile, x staged in LDS
//   - per-lane b128 loads of A feed TWO v_wmma_f32_16x16x4_f32 ops:
//       lanes 0-15  carry K = {k..k+3}   in their (K0,K1) slots,
//       lanes 16-31 carry K = {k+4..k+7} in their (K2,K3) slots,
//     with B supplying the matching hs[] values per slot (K-sum is
//     permutation invariant, so this is exact).
//   - only column N==0 of B is live (lane mask), EXEC stays all-ones
//   - D column 0: lane 0 -> M=0..7 in c[0..7], lane 16 -> M=8..15
//   - xidx != nullptr selects row xidx[0] of xbase (embedding lookup)
// ---------------------------------------------------------------------------
__global__ void k_gemv_wmma(const float* __restrict__ A, const float* __restrict__ xbase,
                            const int* __restrict__ xidx,
                            const float* __restrict__ bias, float* __restrict__ y,
                            int nrows, int K) {
    __shared__ float hs[Hdim];
    const float* x = xidx ? (xbase + (size_t)xidx[0] * K) : xbase;
    for (int i = threadIdx.x; i < K; i += 256) hs[i] = x[i];
    __syncthreads();
    const int wave = threadIdx.x >> 5;
    const int lane = threadIdx.x & 31;
    const int tile = blockIdx.x * 8 + wave;
    const int m    = lane & 15;
    const int hi   = lane >> 4;
    long row = (long)tile * 16 + m;
    if (row >= nrows) row = nrows - 1;                   // clamp loads on tail tile
    const float bm = (m == 0) ? 1.0f : 0.0f;             // only column N==0 live

#if __has_builtin(__builtin_amdgcn_wmma_f32_16x16x4_f32)
    const float* arow = A + row * (long)K;
    v8f c = {};
    for (int k = 0; k < K; k += 8) {
        float4 f = *reinterpret_cast<const float4*>(arow + k + hi * 4);  // b128
        v2f a0 = { f.x, f.y };
        v2f a1 = { f.z, f.w };
        v2f b0 = { bm * hs[k + hi * 4],     bm * hs[k + hi * 4 + 1] };
        v2f b1 = { bm * hs[k + hi * 4 + 2], bm * hs[k + hi * 4 + 3] };
        c = __builtin_amdgcn_wmma_f32_16x16x4_f32(false, a0, false, b0,
                                                  (short)0, c, false, false);
        c = __builtin_amdgcn_wmma_f32_16x16x4_f32(false, a1, false, b1,
                                                  (short)0, c, false, false);
    }
    if (m == 0) {
        long base = (long)tile * 16 + hi * 8;
        for (int r = 0; r < 8; ++r) {
            long idx = base + r;
            if (idx < nrows) y[idx] = c[r] + (bias ? bias[idx] : 0.f);
        }
    }
#else
    const float* arow = A + row * (long)K;               // host-pass / fallback
    float acc = 0.f;
    for (int k = 0; k < K; ++k) acc += arow[k] * hs[k];
    (void)bm;
    long idx = (long)tile * 16 + m;
    if (hi == 0 && idx < nrows) y[idx] = acc + (bias ? bias[idx] : 0.f);
#endif
}

// ---------------------------------------------------------------------------
// softmax over SEQ energies -> attn_weights (written straight to d_out region)
// ---------------------------------------------------------------------------
__global__ void k_softmax(const float* __restrict__ e, float* __restrict__ w) {
    __shared__ float sm[1024];
    int t = threadIdx.x;
    float m = -__FLT_MAX__;
    for (int s = t; s < SEQ; s += 1024) m = fmaxf(m, e[s]);
    sm[t] = m; __syncthreads();
    for (int off = 512; off > 0; off >>= 1) {
        if (t < off) sm[t] = fmaxf(sm[t], sm[t + off]);
        __syncthreads();
    }
    float M = sm[0]; __syncthreads();
    float sum = 0.f;
    for (int s = t; s < SEQ; s += 1024) sum += __expf(e[s] - M);
    sm[t] = sum; __syncthreads();
    for (int off = 512; off > 0; off >>= 1) {
        if (t < off) sm[t] += sm[t + off];
        __syncthreads();
    }
    float inv = 1.0f / sm[0];
    for (int s = t; s < SEQ; s += 1024) w[s] = __expf(e[s] - M) * inv;
}

// ---------------------------------------------------------------------------
// context[h] = sum_s w[s] * enc[s,h]   (enc column reduce; coalesced over h,
// enc is hot in the 192MB L2 after the energies pass)
// ---------------------------------------------------------------------------
__global__ void k_context(const float* __restrict__ enc, const float* __restrict__ w,
                          float* __restrict__ ctx) {
    int h = blockIdx.x * blockDim.x + threadIdx.x;      // 0..Hdim-1
    float acc = 0.f;
    for (int s = 0; s < SEQ; ++s) acc += w[s] * enc[(size_t)s * Hdim + h];
    ctx[h] = acc;
}

// ---------------------------------------------------------------------------
// GRU combine: h_new = (1-z)*n + z*h
// ---------------------------------------------------------------------------
__global__ void k_gru(const float* __restrict__ gi, const float* __restrict__ gh,
                      const float* __restrict__ hprev, float* __restrict__ hnew) {
    int j = blockIdx.x * blockDim.x + threadIdx.x;      // 0..Hdim-1
    float ir = gi[j], iz = gi[Hdim + j], in_ = gi[2 * Hdim + j];
    float hr = gh[j], hz = gh[Hdim + j], hn  = gh[2 * Hdim + j];
    float r = 1.f / (1.f + __expf(-(ir + hr)));
    float z = 1.f / (1.f + __expf(-(iz + hz)));
    float n = tanhf(in_ + r * hn);
    hnew[j] = (1.f - z) * n + z * hprev[j];
}

// ---------------------------------------------------------------------------
// logsumexp over logits -> red[0] = max + log(sum exp(l - max))
// ---------------------------------------------------------------------------
__global__ void k_lse(const float* __restrict__ logits, float* __restrict__ red) {
    __shared__ float sm[1024];
    int t = threadIdx.x;
    float m = -__FLT_MAX__;
    for (int i = t; i < NVOCAB; i += 1024) m = fmaxf(m, logits[i]);
    sm[t] = m; __syncthreads();
    for (int off = 512; off > 0; off >>= 1) {
        if (t < off) sm[t] = fmaxf(sm[t], sm[t + off]);
        __syncthreads();
    }
    float M = sm[0]; __syncthreads();
    float s = 0.f;
    for (int i = t; i < NVOCAB; i += 1024) s += __expf(logits[i] - M);
    sm[t] = s; __syncthreads();
    for (int off = 512; off > 0; off >>= 1) {
        if (t < off) sm[t] += sm[t + off];
        __syncthreads();
    }
    if (t == 0) red[0] = M + logf(sm[0]);
}

__global__ void k_final(const float* __restrict__ red, float* __restrict__ logits) {
    int i = blockIdx.x * blockDim.x + threadIdx.x;
    if (i < NVOCAB) logits[i] -= red[0];
}

// ---------------------------------------------------------------------------
extern "C" void kernel_launch(void* const* d_in, const int* in_sizes, int n_in,
                              void* d_out, int out_size, void* d_ws, size_t ws_size,
                              hipStream_t stream) {
    const int*   word   = (const int*)  d_in[0];
    const float* hprev  = (const float*)d_in[1];
    const float* enc    = (const float*)d_in[2];
    const float* emb    = (const float*)d_in[3];
    const float* attn_W = (const float*)d_in[4];
    // d_in[5] = attn_b: constant shift across s -> exactly cancels in softmax
    const float* v      = (const float*)d_in[6];
    const float* W_ih   = (const float*)d_in[7];
    const float* W_hh   = (const float*)d_in[8];
    const float* b_ih   = (const float*)d_in[9];
    const float* b_hh   = (const float*)d_in[10];
    const float* out_W  = (const float*)d_in[11];
    const float* out_b  = (const float*)d_in[12];

    float* out      = (float*)d_out;
    float* o_logits = out;                               // 50257 (log-softmax)
    float* o_h      = out + NVOCAB;                      // 1024  (h_new)
    float* o_aw     = out + NVOCAB + Hdim;               // 4096  (attn weights)
    float* o_ctx    = out + NVOCAB + Hdim + SEQ;         // 1024  (context)

    float* ws    = (float*)d_ws;
    float* w_u   = ws;                                   // 1024
    float* w_e   = ws + 1024;                            // 4096
    float* w_gi  = ws + 5120;                            // 3072
    float* w_gh  = ws + 8192;                            // 3072
    float* w_red = ws + 11264;                           // 1

    // 1) attention collapse vector
    k_u<<<Hdim / 256, 256, 0, stream>>>(attn_W, v, w_u);

    // 2) energies e[s] = enc[s,:] . u   (4096 rows, K=1024) on WMMA pipe
    k_gemv_wmma<<<SEQ / 128, 256, 0, stream>>>(enc, w_u, nullptr, nullptr,
                                               w_e, SEQ, Hdim);
    // 3) softmax -> attention weights
    k_softmax<<<1, 1024, 0, stream>>>(w_e, o_aw);
    // 4) context
    k_context<<<Hdim / 256, 256, 0, stream>>>(enc, o_aw, o_ctx);

    // 5) GRU gate GEMVs on WMMA pipe
    k_gemv_wmma<<<(3 * Hdim) / 128, 256, 0, stream>>>(W_ih, emb, word, b_ih,
                                                      w_gi, 3 * Hdim, EMBdim);
    k_gemv_wmma<<<(3 * Hdim) / 128, 256, 0, stream>>>(W_hh, hprev, nullptr, b_hh,
                                                      w_gh, 3 * Hdim, Hdim);
    // 6) GRU combine
    k_gru<<<Hdim / 256, 256, 0, stream>>>(w_gi, w_gh, hprev, o_h);

    // 7) logits GEMV (50257 rows, K=1024, 206MB stream) on WMMA pipe
    int tiles  = (NVOCAB + 15) / 16;                     // 3142
    int blocks = (tiles + 7) / 8;                        // 393 (8 waves/block)
    k_gemv_wmma<<<blocks, 256, 0, stream>>>(out_W, o_h, nullptr, out_b,
                                            o_logits, NVOCAB, Hdim);
    // 8) log-softmax
    k_lse<<<1, 1024, 0, stream>>>(o_logits, w_red);
    k_final<<<(NVOCAB + 255) / 256, 256, 0, stream>>>(w_red, o_logits);
}